// GCN_45071386805056
// MI455X (gfx1250) — compile-verified
//
#include <hip/hip_runtime.h>

typedef __attribute__((ext_vector_type(2))) float v2f;
typedef __attribute__((ext_vector_type(8))) float v8f;

#define NNODES 100000
#define NEDGES 3200000
#define NFEAT  512
#define NHID1  16
#define NHID2  64
#define NCLASS 16
#define NEG_SLOPE 0.01f

// ---------------------------------------------------------------------------
// Dense skinny GEMM via V_WMMA_F32_16X16X4_F32 (wave32).
// out[NNODES, NCOL] = act(A [+ bias_in]) @ W[K, NCOL]
//   act = leaky_relu if FUSE_IN (fuses previous layer's bias+activation).
//
// W is staged into LDS once per block (32 KB max, vs 320 KB/WGP), so the
// K-loop issues only the A stream to global memory (the real HBM traffic)
// and reads B from LDS (conflict-free: lanes 0-15 hit banks m, lanes 16-31
// hit banks 32+m).
//
// Each wave computes one 16x16 output tile. Block = 2 row-tiles x (NCOL/16)
// col-waves. 100000 rows = 3125 blocks x 2 tiles x 16 rows exactly, so EXEC
// is all-ones everywhere (WMMA requirement).
//
// VGPR layouts per CDNA5 ISA 7.12.2:
//   A 16x4 f32 : lane m=lane%16 is row; lane<16 holds K={0,1}, lane>=16 K={2,3}
//   B 4x16 f32 : lane n=lane%16 is col; same K split
//   C/D 16x16  : acc[v] = D[v + 8*(lane/16)][lane%16]
// ---------------------------------------------------------------------------
template <int K, int NCOL, bool FUSE_IN>
__global__ __launch_bounds__(64 * (NCOL / 16)) void gemm_wmma(
    const float* __restrict__ A, const float* __restrict__ W,
    const float* __restrict__ bias_in, float* __restrict__ out)
{
    constexpr int CW = NCOL / 16;                  // col waves per block
    constexpr int NTHREADS = 64 * CW;
    constexpr int WELEMS = K * NCOL;               // 8192 / 1024 / 1024 floats

    __shared__ float lw[WELEMS];

    // Cooperative stage of W into LDS (float4 granularity).
    #pragma unroll
    for (int i = threadIdx.x; i < WELEMS / 4; i += NTHREADS)
        ((float4*)lw)[i] = ((const float4*)W)[i];
    __syncthreads();

    const int wave = threadIdx.x >> 5;
    const int lane = threadIdx.x & 31;
    const int col0 = (wave % CW) * 16;
    const int row0 = (blockIdx.x * 2 + wave / CW) * 16;
    const int m  = lane & 15;                      // row (A) / col (B,D)
    const int kg = lane >> 4;                      // K-pair select: 0 or 1

    const float* arow = A + (size_t)(row0 + m) * K;

    v8f acc = {};
    #pragma unroll 4
    for (int k = 0; k < K; k += 4) {
        const int ka = k + kg * 2;
        float a0 = arow[ka];
        float a1 = arow[ka + 1];
        if (FUSE_IN) {
            a0 += bias_in[ka];
            a1 += bias_in[ka + 1];
            a0 = (a0 > 0.f) ? a0 : a0 * NEG_SLOPE;
            a1 = (a1 > 0.f) ? a1 : a1 * NEG_SLOPE;
        }
        v2f av; av[0] = a0; av[1] = a1;
        v2f bv;
        bv[0] = lw[ka * NCOL + col0 + m];
        bv[1] = lw[(ka + 1) * NCOL + col0 + m];
        // 8 args: (neg_a, A, neg_b, B, c_mod, C, reuse_a, reuse_b)
        acc = __builtin_amdgcn_wmma_f32_16x16x4_f32(false, av, false, bv,
                                                    (short)0, acc, false, false);
    }

    #pragma unroll
    for (int v = 0; v < 8; ++v)
        out[(size_t)(row0 + v + 8 * kg) * NCOL + col0 + m] = acc[v];
}

// ---------------------------------------------------------------------------
// SpMM scatter: agg[dst[e], :] += ev[e] * sup[src[e], :]
// One thread per (edge, float4 chunk). sup/agg buffers are L2-resident
// (<= 25.6 MB vs 192 MB L2), so gathers + atomics stay on-chip.
// ---------------------------------------------------------------------------
template <int F>
__global__ void spmm_scatter(const float* __restrict__ sup,
                             const int* __restrict__ src,
                             const int* __restrict__ dst,
                             const float* __restrict__ ev,
                             float* __restrict__ agg)
{
    constexpr int CH = F / 4;                      // float4 chunks per edge
    const unsigned tid = blockIdx.x * blockDim.x + threadIdx.x;
    const unsigned e = tid / CH;
    if (e >= NEDGES) return;
    const unsigned c = (tid % CH) * 4;

    const float  w = ev[e];
    const float4 v = *(const float4*)(sup + (size_t)src[e] * F + c);
    float* o = agg + (size_t)dst[e] * F + c;
    atomicAdd(o + 0, w * v.x);
    atomicAdd(o + 1, w * v.y);
    atomicAdd(o + 2, w * v.z);
    atomicAdd(o + 3, w * v.w);
}

__global__ void zero_f32(float* __restrict__ p, unsigned n)
{
    const unsigned i = blockIdx.x * blockDim.x + threadIdx.x;
    if (i < n) p[i] = 0.f;
}

// Final: out = log_softmax(agg + b3) over 16 classes, one thread per node.
__global__ void bias_logsoftmax16(const float* __restrict__ agg,
                                  const float* __restrict__ b,
                                  float* __restrict__ out)
{
    const unsigned node = blockIdx.x * blockDim.x + threadIdx.x;
    if (node >= NNODES) return;
    float v[NCLASS];
    float mx = -3.402823466e38f;
    #pragma unroll
    for (int j = 0; j < NCLASS; ++j) {
        v[j] = agg[(size_t)node * NCLASS + j] + b[j];
        mx = fmaxf(mx, v[j]);
    }
    float s = 0.f;
    #pragma unroll
    for (int j = 0; j < NCLASS; ++j) s += __expf(v[j] - mx);
    const float lse = mx + __logf(s);
    #pragma unroll
    for (int j = 0; j < NCLASS; ++j)
        out[(size_t)node * NCLASS + j] = v[j] - lse;
}

// ---------------------------------------------------------------------------
extern "C" void kernel_launch(void* const* d_in, const int* in_sizes, int n_in,
                              void* d_out, int out_size, void* d_ws, size_t ws_size,
                              hipStream_t stream)
{
    const float* x  = (const float*)d_in[0];
    const int*   sr = (const int*)  d_in[1];
    const int*   ds = (const int*)  d_in[2];
    const float* ev = (const float*)d_in[3];
    const float* W1 = (const float*)d_in[4];
    const float* b1 = (const float*)d_in[5];
    const float* W2 = (const float*)d_in[6];
    const float* b2 = (const float*)d_in[7];
    const float* W3 = (const float*)d_in[8];
    const float* b3 = (const float*)d_in[9];

    float* ws = (float*)d_ws;                       // 64 MB used total
    float* s1 = ws;                                 // [100000,16]
    float* a1 = s1 + (size_t)NNODES * NHID1;        // [100000,16]
    float* s2 = a1 + (size_t)NNODES * NHID1;        // [100000,64]
    float* a2 = s2 + (size_t)NNODES * NHID2;        // [100000,64]
    // layer 3 reuses s1 (support) and a1 (agg)

    const int rowBlocks = NNODES / 32;              // 3125: 2 row-tiles/block

    // Layer 1: support1 = x @ W1 ; agg1 = spmm(support1)
    gemm_wmma<NFEAT, NHID1, false><<<rowBlocks, 64, 0, stream>>>(x, W1, nullptr, s1);
    zero_f32<<<(NNODES * NHID1) / 256, 256, 0, stream>>>(a1, NNODES * NHID1);
    spmm_scatter<NHID1><<<(NEDGES * (NHID1 / 4)) / 256, 256, 0, stream>>>(s1, sr, ds, ev, a1);

    // Layer 2: support2 = leaky(agg1 + b1) @ W2 ; agg2 = spmm(support2)
    gemm_wmma<NHID1, NHID2, true><<<rowBlocks, 256, 0, stream>>>(a1, W2, b1, s2);
    zero_f32<<<(NNODES * NHID2) / 256, 256, 0, stream>>>(a2, NNODES * NHID2);
    spmm_scatter<NHID2><<<(NEDGES * (NHID2 / 4)) / 256, 256, 0, stream>>>(s2, sr, ds, ev, a2);

    // Layer 3: support3 = leaky(agg2 + b2) @ W3 ; agg3 = spmm(support3)
    gemm_wmma<NHID2, NCLASS, true><<<rowBlocks, 64, 0, stream>>>(a2, W3, b2, s1);
    zero_f32<<<(NNODES * NCLASS) / 256, 256, 0, stream>>>(a1, NNODES * NCLASS);
    spmm_scatter<NCLASS><<<(NEDGES * (NCLASS / 4)) / 256, 256, 0, stream>>>(s1, sr, ds, ev, a1);

    // out = log_softmax(agg3 + b3)
    bias_logsoftmax16<<<(NNODES + 255) / 256, 256, 0, stream>>>(a1, b3, (float*)d_out);
}